// Net_11940009083302
// MI455X (gfx1250) — compile-verified
//
#include <hip/hip_runtime.h>
#include <cstdint>
#include <cstddef>

// ============================================================================
// PointTransformer forward for MI455X (gfx1250), fp32 end-to-end.
// All matmuls -> V_WMMA_F32_16X16X4_F32 (wave32, 16x16 tile per wave),
// 4 waves/block sharing an LDS-staged weight tile, K unrolled x4 so the
// interior loop is branch-free with 4 back-to-back WMMAs per 16-K chunk.
// ============================================================================

#define BATCH 8
#define N0_   4096
#define KNN_  16
#define EPSF  1e-5f
#define GEMM_WAVES 4

typedef __attribute__((ext_vector_type(2))) float v2f;
typedef __attribute__((ext_vector_type(8))) float v8f;

static inline int cdiv_h(int a, int b){ return (a + b - 1) / b; }

// ---------------------------------------------------------------------------
// Generic GEMM: C[M,N] = A[M,Kd] @ W[Kd,N] (+bias) (+relu), row-major.
// Block = 128 threads = 4 waves; wave w owns M-tile (blockIdx.x*4+w)*16,
// all waves share N-tile blockIdx.y*16 whose weight chunk lives in LDS.
// ---------------------------------------------------------------------------
__global__ void k_gemm(const float* __restrict__ A, const float* __restrict__ W,
                       const float* __restrict__ bias, float* __restrict__ C,
                       int M, int N, int Kd, int relu)
{
  __shared__ float bs[16][17];               // [k_local][col], padded
  const int tid  = threadIdx.x;              // 0..127
  const int wave = tid >> 5;
  const int lane = tid & 31;
  const int half = lane >> 4;
  const int l16  = lane & 15;
  const int row0 = (blockIdx.x * GEMM_WAVES + wave) * 16;
  const int col0 = blockIdx.y * 16;
  const int arow = row0 + l16;
  const bool fastrow = (row0 + 16 <= M);

  v8f acc = {0.f,0.f,0.f,0.f,0.f,0.f,0.f,0.f};

  for (int k0 = 0; k0 < Kd; k0 += 16) {
    // ---- cooperative load of weight chunk W[k0..k0+15][col0..col0+15] ----
    {
      int kk = tid >> 4, cc = tid & 15;                     // element tid
      float v = 0.f;
      if (k0 + kk < Kd && col0 + cc < N) v = W[(size_t)(k0 + kk) * N + col0 + cc];
      bs[kk][cc] = v;
      kk = (tid + 128) >> 4; cc = tid & 15;                 // element tid+128
      v = 0.f;
      if (k0 + kk < Kd && col0 + cc < N) v = W[(size_t)(k0 + kk) * N + col0 + cc];
      bs[kk][cc] = v;
    }
    __syncthreads();

    if (fastrow && (Kd - k0) >= 16) {
      // ---- branch-free interior path: vector A loads, 4 WMMAs ----
      const float* ap = A + (size_t)arow * Kd + k0 + half * 2;
      if (k0 + 16 < Kd) __builtin_prefetch(ap + 16, 0, 1);  // global_prefetch_b8
#pragma unroll
      for (int s = 0; s < 4; ++s) {
        const v2f av = *(const v2f*)(ap + s * 4);
        v2f bv; bv.x = bs[s*4 + half*2][l16]; bv.y = bs[s*4 + half*2 + 1][l16];
        acc = __builtin_amdgcn_wmma_f32_16x16x4_f32(false, av, false, bv,
                                                    (short)0, acc, false, false);
      }
    } else {
      // ---- edge path (M tail or Kd in {3,6}): guarded scalar A loads ----
#pragma unroll
      for (int s = 0; s < 4; ++s) {
        const int ka = k0 + s * 4 + half * 2;
        float a0 = 0.f, a1 = 0.f;
        if (arow < M) {
          if (ka     < Kd) a0 = A[(size_t)arow * Kd + ka];
          if (ka + 1 < Kd) a1 = A[(size_t)arow * Kd + ka + 1];
        }
        v2f av; av.x = a0; av.y = a1;
        v2f bv; bv.x = bs[s*4 + half*2][l16]; bv.y = bs[s*4 + half*2 + 1][l16];
        acc = __builtin_amdgcn_wmma_f32_16x16x4_f32(false, av, false, bv,
                                                    (short)0, acc, false, false);
      }
    }
    __syncthreads();
  }

#pragma unroll
  for (int i = 0; i < 8; ++i) {
    const int r = row0 + half * 8 + i;
    const int c = col0 + l16;
    if (r < M && c < N) {
      float v = acc[i];
      if (bias) v += bias[c];
      if (relu) v = fmaxf(v, 0.f);
      C[(size_t)r * N + c] = v;
    }
  }
}

// ----------------------------- BatchNorm -----------------------------------
__global__ void k_bn_stats(const float* __restrict__ x, int N, int C,
                           float* __restrict__ mu, float* __restrict__ var)
{
  __shared__ float s1[256], s2[256];
  const int c = blockIdx.x;
  float a = 0.f, b = 0.f;
  for (int i = threadIdx.x; i < N; i += blockDim.x) {
    const float v = x[(size_t)i * C + c];
    a += v; b += v * v;
  }
  s1[threadIdx.x] = a; s2[threadIdx.x] = b;
  __syncthreads();
  for (int s = 128; s > 0; s >>= 1) {
    if ((int)threadIdx.x < s) { s1[threadIdx.x] += s1[threadIdx.x + s];
                                s2[threadIdx.x] += s2[threadIdx.x + s]; }
    __syncthreads();
  }
  if (threadIdx.x == 0) {
    const float m = s1[0] / (float)N;
    mu[c] = m;
    var[c] = s2[0] / (float)N - m * m;   // population var == jnp.var
  }
}

__global__ void k_bn_relu(const float* __restrict__ x, const float* __restrict__ mu,
                          const float* __restrict__ var, const float* __restrict__ g,
                          const float* __restrict__ be, float* __restrict__ y,
                          int N, int C)
{
  const int i = blockIdx.x * blockDim.x + threadIdx.x;
  if (i >= N * C) return;
  const int c = i % C;
  const float s = g[c] * rsqrtf(var[c] + EPSF);
  const float v = (x[i] - mu[c]) * s + be[c];
  y[i] = fmaxf(v, 0.f);
}

// ----------------------------- kNN -----------------------------------------
// One thread per query: 16-entry sorted insertion list over all nk keys.
// excl==1 -> self excluded (knn_graph); bi init = qi reproduces the reference
// n==k corner case (self re-appears with inf distance at the 16-point level).
__global__ void k_knn(const float* __restrict__ kpos, int nk,
                      const float* __restrict__ qpos, int nq,
                      int* __restrict__ nbr, int excl)
{
  const int t = blockIdx.x * blockDim.x + threadIdx.x;
  if (t >= BATCH * nq) return;
  const int b = t / nq, qi = t % nq;
  const float* kp = kpos + (size_t)b * nk * 3;
  const float* qp = qpos + (size_t)t * 3;
  const float qx = qp[0], qy = qp[1], qz = qp[2];

  float bd[KNN_]; int bi[KNN_];
  for (int j = 0; j < KNN_; ++j) { bd[j] = 3.4e38f; bi[j] = qi; }

  for (int i = 0; i < nk; ++i) {
    if (excl && i == qi) continue;
    const float dx = kp[i*3] - qx, dy = kp[i*3+1] - qy, dz = kp[i*3+2] - qz;
    const float d = dx*dx + dy*dy + dz*dz;
    if (d < bd[KNN_-1]) {
      int j = KNN_ - 1;
      while (j > 0 && bd[j-1] > d) { bd[j] = bd[j-1]; bi[j] = bi[j-1]; --j; }
      bd[j] = d; bi[j] = i;
    }
  }
  for (int j = 0; j < KNN_; ++j) nbr[(size_t)t * KNN_ + j] = bi[j];
}

// ----------------------------- FPS ------------------------------------------
// One block (256 threads) per cloud; dmin in LDS; first-index tie-break argmax.
__global__ void k_fps(const float* __restrict__ pos, int n, int m, int* __restrict__ idx)
{
  __shared__ float dmin[4096];
  __shared__ float rv[256];
  __shared__ int   ri[256];
  const float* p = pos + (size_t)blockIdx.x * n * 3;
  int* out = idx + (size_t)blockIdx.x * m;

  for (int i = threadIdx.x; i < n; i += blockDim.x) dmin[i] = 3.4e38f;
  if (threadIdx.x == 0) out[0] = 0;
  __syncthreads();

  int last = 0;
  for (int s = 1; s < m; ++s) {
    const float lx = p[last*3], ly = p[last*3+1], lz = p[last*3+2];
    float bv = -1.f; int bim = 0x7fffffff;
    for (int i = threadIdx.x; i < n; i += blockDim.x) {
      const float dx = p[i*3] - lx, dy = p[i*3+1] - ly, dz = p[i*3+2] - lz;
      const float d  = dx*dx + dy*dy + dz*dz;
      const float dm = fminf(dmin[i], d);
      dmin[i] = dm;
      if (dm > bv) { bv = dm; bim = i; }   // strided i ascending -> first max
    }
    rv[threadIdx.x] = bv; ri[threadIdx.x] = bim;
    __syncthreads();
    for (int st = 128; st > 0; st >>= 1) {
      if ((int)threadIdx.x < st) {
        const float v2 = rv[threadIdx.x + st]; const int i2 = ri[threadIdx.x + st];
        if (v2 > rv[threadIdx.x] || (v2 == rv[threadIdx.x] && i2 < ri[threadIdx.x])) {
          rv[threadIdx.x] = v2; ri[threadIdx.x] = i2;
        }
      }
      __syncthreads();
    }
    last = ri[0];
    __syncthreads();
    if (threadIdx.x == 0) out[s] = last;
  }
}

// ----------------------------- small gather / misc --------------------------
__global__ void k_gather3(const float* __restrict__ pos, const int* __restrict__ idx,
                          float* __restrict__ out, int n, int m)
{
  const int t = blockIdx.x * blockDim.x + threadIdx.x;
  if (t >= BATCH * m) return;
  const int b = t / m;
  const int src = idx[t];
  const float* p = pos + ((size_t)b * n + src) * 3;
  out[(size_t)t*3+0] = p[0]; out[(size_t)t*3+1] = p[1]; out[(size_t)t*3+2] = p[2];
}

__global__ void k_nbr_max(const float* __restrict__ h, const int* __restrict__ nbr,
                          float* __restrict__ out, int n, int m, int C)
{
  const int t = blockIdx.x * blockDim.x + threadIdx.x;
  if (t >= BATCH * m * C) return;
  const int c = t % C; const int r = t / C;
  const int b = r / m; const int j = r % m;
  const int* nb = nbr + ((size_t)b * m + j) * KNN_;
  float v = -3.4e38f;
  for (int k = 0; k < KNN_; ++k)
    v = fmaxf(v, h[((size_t)b * n + nb[k]) * C + c]);
  out[t] = v;
}

// rel[e,:] = pos[target] - pos[neighbor]; slot 16 = self-loop.
__global__ void k_rel(const float* __restrict__ pos, const int* __restrict__ nbr,
                      float* __restrict__ rel, int n)
{
  const int e = blockIdx.x * blockDim.x + threadIdx.x;
  if (e >= n * 17) return;
  const int qi = e / 17, s = e % 17;
  const int nb = (s < KNN_) ? nbr[qi * KNN_ + s] : qi;
  rel[e*3+0] = pos[qi*3+0] - pos[nb*3+0];
  rel[e*3+1] = pos[qi*3+1] - pos[nb*3+1];
  rel[e*3+2] = pos[qi*3+2] - pos[nb*3+2];
}

// tdiff = xd[target] - xs[neighbor] + delta   (attn_nn input)
__global__ void k_tdiff(const float* __restrict__ xd, const float* __restrict__ xs,
                        const int* __restrict__ nbr, const float* __restrict__ delta,
                        float* __restrict__ out, int n, int C)
{
  const int t = blockIdx.x * blockDim.x + threadIdx.x;
  if (t >= n * 17 * C) return;
  const int c = t % C; const int e = t / C;
  const int qi = e / 17, s = e % 17;
  const int nb = (s < KNN_) ? nbr[qi * KNN_ + s] : qi;
  out[t] = xd[(size_t)qi * C + c] - xs[(size_t)nb * C + c] + delta[t];
}

// per (target, channel): softmax over 17 slots, weighted sum of (xv[nbr]+delta)
__global__ void k_aggregate(const float* __restrict__ a, const float* __restrict__ xv,
                            const float* __restrict__ delta, const int* __restrict__ nbr,
                            float* __restrict__ out, int n, int C)
{
  const int t = blockIdx.x * blockDim.x + threadIdx.x;
  if (t >= n * C) return;
  const int qi = t / C, c = t % C;
  float mx = -3.4e38f;
  for (int s = 0; s < 17; ++s)
    mx = fmaxf(mx, a[((size_t)qi * 17 + s) * C + c]);
  float den = 0.f, num = 0.f;
  for (int s = 0; s < 17; ++s) {
    const float w = expf(a[((size_t)qi * 17 + s) * C + c] - mx);
    const int nb = (s < KNN_) ? nbr[qi * KNN_ + s] : qi;
    num += w * (xv[(size_t)nb * C + c] + delta[((size_t)qi * 17 + s) * C + c]);
    den += w;
  }
  out[t] = num / den;
}

__global__ void k_mean(const float* __restrict__ x, float* __restrict__ g, int n, int C)
{
  const int t = blockIdx.x * blockDim.x + threadIdx.x;
  if (t >= BATCH * C) return;
  const int b = t / C, c = t % C;
  float s = 0.f;
  for (int i = 0; i < n; ++i) s += x[((size_t)b * n + i) * C + c];
  g[t] = s / (float)n;
}

__global__ void k_lsm(const float* __restrict__ in, float* __restrict__ out,
                      int rows, int cols)
{
  const int r = blockIdx.x * blockDim.x + threadIdx.x;
  if (r >= rows) return;
  float mx = -3.4e38f;
  for (int c = 0; c < cols; ++c) mx = fmaxf(mx, in[(size_t)r * cols + c]);
  float s = 0.f;
  for (int c = 0; c < cols; ++c) s += expf(in[(size_t)r * cols + c] - mx);
  const float l = logf(s) + mx;
  for (int c = 0; c < cols; ++c) out[(size_t)r * cols + c] = in[(size_t)r * cols + c] - l;
}

// ============================================================================
// Host orchestration
// ============================================================================
struct TBP {
  const float *a0b,*a0w,*a1b,*a1w,*linw,*dstw,*inb,*inw,*outb,*outw,*srcw,
              *p0b,*p0w,*p1b,*p1w;
};

static void gemm(const float* A, const float* W, const float* bias, float* C,
                 int M, int N, int Kd, int relu, hipStream_t s)
{
  dim3 g(cdiv_h(M, 16 * GEMM_WAVES), cdiv_h(N, 16));
  k_gemm<<<g, dim3(32 * GEMM_WAVES), 0, s>>>(A, W, bias, C, M, N, Kd, relu);
}

static void run_tb(const float* x, const float* pos, const int* nbr, const TBP& P,
                   int n, int C,
                   float* t1, float* t2, float* t3, float* t4, float* t5,
                   float* e_rel, float* e_hid, float* e_delta, float* e_tdiff,
                   float* e_attn, float* out, hipStream_t s)
{
  const int N = BATCH * n;
  gemm(x,  P.inw,  P.inb, t1, N, C, C, 1, s);          // x1 = relu(lin_in(x))
  gemm(t1, P.srcw, nullptr, t2, N, C, C, 0, s);        // xs
  gemm(t1, P.dstw, nullptr, t3, N, C, C, 0, s);        // xd
  gemm(t1, P.linw, nullptr, t4, N, C, C, 0, s);        // xv
  const int E = n * 17;
  for (int b = 0; b < BATCH; ++b) {
    const float* pb = pos + (size_t)b * n * 3;
    const int*   nb = nbr + (size_t)b * n * KNN_;
    k_rel<<<cdiv_h(E, 256), 256, 0, s>>>(pb, nb, e_rel, n);
    gemm(e_rel, P.p0w, P.p0b, e_hid,   E, 64, 3,  1, s);   // pos_nn layer 1
    gemm(e_hid, P.p1w, P.p1b, e_delta, E, C,  64, 1, s);   // pos_nn layer 2
    k_tdiff<<<cdiv_h(E * C, 256), 256, 0, s>>>(t3 + (size_t)b*n*C, t2 + (size_t)b*n*C,
                                               nb, e_delta, e_tdiff, n, C);
    gemm(e_tdiff, P.a0w, P.a0b, e_hid,  E, 64, C,  1, s);  // attn_nn layer 1
    gemm(e_hid,   P.a1w, P.a1b, e_attn, E, C,  64, 1, s);  // attn_nn layer 2
    k_aggregate<<<cdiv_h(n * C, 256), 256, 0, s>>>(e_attn, t4 + (size_t)b*n*C,
                                                   e_delta, nb, t5 + (size_t)b*n*C, n, C);
  }
  gemm(t5, P.outw, P.outb, out, N, C, C, 1, s);        // relu(lin_out(out))
}

extern "C" void kernel_launch(void* const* d_in, const int* in_sizes, int n_in,
                              void* d_out, int out_size, void* d_ws, size_t ws_size,
                              hipStream_t stream)
{
  (void)n_in; (void)out_size; (void)ws_size;
  static const int DIMS_[5] = {32, 64, 128, 256, 512};

  const float* x_in   = (const float*)d_in[0];
  const float* pos_in = (const float*)d_in[1];
  // d_in[2] = batch (unused; equal-sized clouds)

  // ---- parameter pointer mapping -----------------------------------------
  // Primary: top-level insertion order (x,pos,batch,params), params flattened
  // in JAX tree (alphabetical-key) order. Fallback (in_sizes[3]==192):
  // params flattened in dict insertion order.
  const bool ins = (in_sizes[3] == 192);
  auto F = [&](int i)->const float* { return (const float*)d_in[i]; };

  const float *mlp_w, *mlp_b, *mlp_g, *mlp_be;
  const float *dn_w[4], *dn_b[4], *dn_g[4], *dn_be[4];
  const float *o1_w, *o1_b, *o2_w, *o2_b;
  TBP tbin, tbd[4];

  auto tb_sorted = [&](int base)->TBP {
    TBP t; t.a0b=F(base+0); t.a0w=F(base+1); t.a1b=F(base+2); t.a1w=F(base+3);
    t.linw=F(base+4); t.dstw=F(base+5); t.inb=F(base+6); t.inw=F(base+7);
    t.outb=F(base+8); t.outw=F(base+9); t.srcw=F(base+10);
    t.p0b=F(base+11); t.p0w=F(base+12); t.p1b=F(base+13); t.p1w=F(base+14);
    return t;
  };
  auto tb_insert = [&](int base)->TBP {
    TBP t; t.inw=F(base+0); t.inb=F(base+1); t.outw=F(base+2); t.outb=F(base+3);
    t.p0w=F(base+4); t.p0b=F(base+5); t.p1w=F(base+6); t.p1b=F(base+7);
    t.a0w=F(base+8); t.a0b=F(base+9); t.a1w=F(base+10); t.a1b=F(base+11);
    t.linw=F(base+12); t.srcw=F(base+13); t.dstw=F(base+14);
    return t;
  };

  if (!ins) {
    for (int i = 0; i < 4; ++i) {
      dn_be[i]=F(3+4*i); dn_g[i]=F(4+4*i); dn_b[i]=F(5+4*i); dn_w[i]=F(6+4*i);
    }
    mlp_be=F(19); mlp_g=F(20); mlp_b=F(21); mlp_w=F(22);
    o1_b=F(23); o1_w=F(24); o2_b=F(25); o2_w=F(26);
    for (int i = 0; i < 4; ++i) tbd[i] = tb_sorted(27 + 15*i);
    tbin = tb_sorted(87);
  } else {
    mlp_w=F(3); mlp_b=F(4); mlp_g=F(5); mlp_be=F(6);
    tbin = tb_insert(7);
    for (int i = 0; i < 4; ++i) {
      dn_w[i]=F(22+4*i); dn_b[i]=F(23+4*i); dn_g[i]=F(24+4*i); dn_be[i]=F(25+4*i);
    }
    for (int i = 0; i < 4; ++i) tbd[i] = tb_insert(38 + 15*i);
    o1_w=F(98); o1_b=F(99); o2_w=F(100); o2_b=F(101);
  }

  // ---- workspace bump allocator (~105 MB peak) ---------------------------
  size_t off = 0;
  auto WA = [&](size_t nf)->float* {
    float* p = (float*)((char*)d_ws + off);
    off += ((nf * sizeof(float)) + 255) & ~(size_t)255;
    return p;
  };
  auto WI = [&](size_t ni)->int* {
    int* p = (int*)((char*)d_ws + off);
    off += ((ni * sizeof(int)) + 255) & ~(size_t)255;
    return p;
  };

  const size_t NCMAX = (size_t)32768 * 64;   // biggest node tensor [B*N0, 64]
  float* t1 = WA(NCMAX); float* t2 = WA(NCMAX); float* t3 = WA(NCMAX);
  float* t4 = WA(NCMAX); float* t5 = WA(NCMAX);
  float* xA = WA(NCMAX); float* xB = WA(NCMAX);
  float* posA = WA((size_t)32768 * 3); float* posB = WA((size_t)32768 * 3);
  float* e_rel   = WA((size_t)N0_ * 17 * 3);
  float* e_hid   = WA((size_t)N0_ * 17 * 64);
  float* e_delta = WA((size_t)N0_ * 17 * 32);
  float* e_tdiff = WA((size_t)N0_ * 17 * 32);
  float* e_attn  = WA((size_t)N0_ * 17 * 32);
  float* mu  = WA(512); float* var = WA(512);
  float* gbuf = WA((size_t)BATCH * 512);
  float* hbuf = WA((size_t)BATCH * 64);
  float* lbuf = WA((size_t)BATCH * 40);
  int* nbr    = WI((size_t)32768 * KNN_);
  int* nbr2   = WI((size_t)BATCH * 1024 * KNN_);
  int* fpsidx = WI((size_t)BATCH * 1024);

  // ---- level 0: input MLP + BN + transformer block -----------------------
  int n = N0_;
  const float* posCur = pos_in;
  float* xa = xA; float* xb = xB;

  gemm(x_in, mlp_w, mlp_b, t1, BATCH * n, 32, 6, 0, stream);
  k_bn_stats<<<32, 256, 0, stream>>>(t1, BATCH * n, 32, mu, var);
  k_bn_relu<<<cdiv_h(BATCH * n * 32, 256), 256, 0, stream>>>(t1, mu, var, mlp_g, mlp_be,
                                                             xa, BATCH * n, 32);
  k_knn<<<cdiv_h(BATCH * n, 128), 128, 0, stream>>>(posCur, n, posCur, n, nbr, 1);
  run_tb(xa, posCur, nbr, tbin, n, 32, t1, t2, t3, t4, t5,
         e_rel, e_hid, e_delta, e_tdiff, e_attn, xb, stream);
  { float* tmp = xa; xa = xb; xb = tmp; }

  // ---- 4 x (transition down + transformer block) -------------------------
  for (int i = 0; i < 4; ++i) {
    const int Cin = DIMS_[i], Cout = DIMS_[i + 1];
    const int m = n / 4;
    // h = relu(bn(lin(x)))  over ALL current nodes
    gemm(xa, dn_w[i], dn_b[i], t1, BATCH * n, Cout, Cin, 0, stream);
    k_bn_stats<<<Cout, 256, 0, stream>>>(t1, BATCH * n, Cout, mu, var);
    k_bn_relu<<<cdiv_h(BATCH * n * Cout, 256), 256, 0, stream>>>(t1, mu, var, dn_g[i],
                                                                 dn_be[i], t2, BATCH * n, Cout);
    // FPS subsample, kNN from sub to full, neighbor-max pool
    k_fps<<<BATCH, 256, 0, stream>>>(posCur, n, m, fpsidx);
    float* posNext = (i & 1) ? posB : posA;
    k_gather3<<<cdiv_h(BATCH * m, 256), 256, 0, stream>>>(posCur, fpsidx, posNext, n, m);
    k_knn<<<cdiv_h(BATCH * m, 128), 128, 0, stream>>>(posCur, n, posNext, m, nbr2, 0);
    k_nbr_max<<<cdiv_h(BATCH * m * Cout, 256), 256, 0, stream>>>(t2, nbr2, xb, n, m, Cout);
    { float* tmp = xa; xa = xb; xb = tmp; }
    posCur = posNext; n = m;
    // kNN graph at the new level + transformer block
    k_knn<<<cdiv_h(BATCH * n, 128), 128, 0, stream>>>(posCur, n, posCur, n, nbr, 1);
    run_tb(xa, posCur, nbr, tbd[i], n, Cout, t1, t2, t3, t4, t5,
           e_rel, e_hid, e_delta, e_tdiff, e_attn, xb, stream);
    { float* tmp = xa; xa = xb; xb = tmp; }
  }

  // ---- head: mean pool -> 512->64 relu -> 64->40 -> log_softmax ----------
  k_mean<<<cdiv_h(BATCH * 512, 256), 256, 0, stream>>>(xa, gbuf, n, 512);
  gemm(gbuf, o1_w, o1_b, hbuf, BATCH, 64, 512, 1, stream);
  gemm(hbuf, o2_w, o2_b, lbuf, BATCH, 40, 64, 0, stream);
  k_lsm<<<1, 64, 0, stream>>>(lbuf, (float*)d_out, BATCH, 40);
}